// Triplet_Distillation_MMD_29970281791726
// MI455X (gfx1250) — compile-verified
//
#include <hip/hip_runtime.h>

typedef float v2f __attribute__((ext_vector_type(2)));
typedef float v4f __attribute__((ext_vector_type(4)));
typedef float v8f __attribute__((ext_vector_type(8)));

#define T_LEN     512
#define DF        64
#define ROW_ELEMS (T_LEN * DF)   // 32768 floats per (b) row
#define NWK       4
#define STEPL     16
#define NSUB      11             // rows per group in MMD
#define NROWS     22             // 2*NSUB
#define NPAIRS    9

// ---------------- block reduction helper ----------------
__device__ __forceinline__ float block_reduce(float v, float* red) {
  int tid = threadIdx.x;
  red[tid] = v;
  __syncthreads();
  for (int s = blockDim.x >> 1; s > 0; s >>= 1) {
    if (tid < s) red[tid] += red[tid + s];
    __syncthreads();
  }
  float r = red[0];
  __syncthreads();
  return r;
}

// ---------------- Kernel A: softDTW (band==diagonal) => sum_t ||x_t - y_t||^2 ----------------
// BANDWIDTH=0.1 with integer |i-j| means only the diagonal is in-band, and
// exp(-INF/GAMMA) underflows to exactly 0, so softDTW == sum of diagonal sq-dists.
__global__ void diag_dtw_kernel(const float* __restrict__ data,
                                const float* __restrict__ teacher,
                                float* __restrict__ sd) {
  __shared__ float red[256];
  int p = blockIdx.x;                  // 0..119
  int q = p % 60;
  const float* src = (p < 60) ? data : teacher;
  int w = q / 15, f = q % 15;
  const v4f* A = (const v4f*)(src + (size_t)(w * STEPL) * ROW_ELEMS);
  const v4f* O = (const v4f*)(src + (size_t)(w * STEPL + 1 + f) * ROW_ELEMS);
  float s = 0.f;
  for (int e = threadIdx.x; e < ROW_ELEMS / 4; e += blockDim.x) {
    v4f dx = A[e] - O[e];
    s += dx.x * dx.x + dx.y * dx.y + dx.z * dx.z + dx.w * dx.w;
  }
  float tot = block_reduce(s, red);
  if (threadIdx.x == 0) sd[p] = tot;
}

// ---------------- Kernel B: multi-kernel MMD per pair, Gram via fp32 WMMA ----------------
// One block (128 thr = 4 wave32) per pair. Wave w owns 16x16 Gram tile (ti=w&1, tj=w>>1)
// of the zero-padded 32x32 Gram. Padded rows/cols load garbage from row 0 instead of
// zeros -- harmless, since Gram rows>=22 / cols>=22 are never consumed.
// Per (row,t) we load float4 of features d..d+3 and drive 4 Gram accumulators, so the
// k-loop issues 4 global_load_b128 : 4 v_wmma_f32_16x16x4_f32.
// bw is computed Gram-free: sum(L2) = 44*sum(sq) - 2*sum_{t,d}(colsum)^2.
__global__ void mmd_kernel(const float* __restrict__ data,
                           float* __restrict__ mmd_out) {
  __shared__ float sq[NROWS * DF];
  __shared__ float red[128];
  __shared__ float inv_sh[5];          // 1/(bw * 2^ii), block-uniform

  const int PI[NPAIRS] = {0,0,0,1,1,2,2,3,3};
  const int PJ[NPAIRS] = {1,2,3,2,3,1,3,1,2};
  int b  = blockIdx.x;
  int gi = PI[b], gj = PJ[b];
  int tid = threadIdx.x;

  auto rowOff = [&](int r) -> size_t {
    int g = (r < NSUB) ? gi : gj;
    int s = (r < NSUB) ? r : r - NSUB;
    return (size_t)(g * STEPL + s) * ROW_ELEMS;
  };

  // ---- step 1: sq[a][d] = sum_t x^2 (coalesced across lanes in d) ----
  float partSq = 0.f;
  for (int e = tid; e < NROWS * DF; e += blockDim.x) {
    int a = e >> 6, d = e & 63;
    const float* base = data + rowOff(a) + d;
    float s = 0.f;
    for (int t = 0; t < T_LEN; ++t) {
      float v = base[(size_t)t * DF];
      s += v * v;
    }
    sq[e] = s;
    partSq += s;
  }
  __syncthreads();

  // ---- step 2: sum over (t,d) of (column-sum over 22 rows)^2, b128 loads ----
  float partCS = 0.f;
  for (int e4 = tid; e4 < ROW_ELEMS / 4; e4 += blockDim.x) {
    v4f s = {0.f, 0.f, 0.f, 0.f};
    for (int r = 0; r < NROWS; ++r)
      s += *(const v4f*)(data + rowOff(r) + (size_t)e4 * 4);
    partCS += s.x * s.x + s.y * s.y + s.z * s.z + s.w * s.w;
  }
  float sumSq = block_reduce(partSq, red);
  float sumCS = block_reduce(partCS, red);
  if (tid == 0) {
    float bwsum = 44.f * sumSq - 2.f * sumCS;      // sum(L2) over 22x22x64
    float bw = bwsum / 462.f / 4.f;                // /(ns^2-ns), / K_MUL^(K_NUM//2)
    float inv = 1.f / bw;
    #pragma unroll
    for (int ii = 0; ii < 5; ++ii) { inv_sh[ii] = inv; inv *= 0.5f; }
  }
  __syncthreads();
  float inv0 = inv_sh[0], inv1 = inv_sh[1], inv2 = inv_sh[2],
        inv3 = inv_sh[3], inv4 = inv_sh[4];

  // ---- step 3: WMMA Gram, fused L2 -> 5-kernel RBF -> signed accumulation ----
  int lane  = tid & 31;
  int wave  = tid >> 5;
  int ti    = wave & 1;
  int tj    = wave >> 1;
  int mReal = ti * 16 + (lane & 15);   // A-matrix row this lane supplies
  int nReal = tj * 16 + (lane & 15);   // B-matrix col this lane supplies
  int kHalf = (lane >> 4) << 1;        // lanes 0-15 -> K {0,1}; 16-31 -> K {2,3}
  int mRow  = (mReal < NROWS) ? mReal : 0;   // clamp: garbage pollutes only dead tiles
  int nRow  = (nReal < NROWS) ? nReal : 0;
  const float* mBase = data + rowOff(mRow);
  const float* nBase = data + rowOff(nRow);
  int colB  = tj * 16 + (lane & 15);         // C/D column this lane holds
  int rBase = ti * 16 + ((lane >> 4) << 3);  // C/D row = rBase + r

  float part = 0.f;
  for (int d0 = 0; d0 < DF; d0 += 4) {
    const float* mp  = mBase + d0;
    const float* np2 = nBase + d0;
    v8f acc0 = {}, acc1 = {}, acc2 = {}, acc3 = {};
    for (int t0 = 0; t0 < T_LEN; t0 += 4) {
      size_t t = (size_t)(t0 + kHalf);
      v4f aLo = *(const v4f*)(mp  + t * DF);
      v4f aHi = *(const v4f*)(mp  + (t + 1) * DF);
      v4f bLo = *(const v4f*)(np2 + t * DF);
      v4f bHi = *(const v4f*)(np2 + (t + 1) * DF);
      v2f af, bf;
      af.x = aLo.x; af.y = aHi.x; bf.x = bLo.x; bf.y = bHi.x;
      acc0 = __builtin_amdgcn_wmma_f32_16x16x4_f32(false, af, false, bf, (short)0, acc0, false, false);
      af.x = aLo.y; af.y = aHi.y; bf.x = bLo.y; bf.y = bHi.y;
      acc1 = __builtin_amdgcn_wmma_f32_16x16x4_f32(false, af, false, bf, (short)0, acc1, false, false);
      af.x = aLo.z; af.y = aHi.z; bf.x = bLo.z; bf.y = bHi.z;
      acc2 = __builtin_amdgcn_wmma_f32_16x16x4_f32(false, af, false, bf, (short)0, acc2, false, false);
      af.x = aLo.w; af.y = aHi.w; bf.x = bLo.w; bf.y = bHi.w;
      acc3 = __builtin_amdgcn_wmma_f32_16x16x4_f32(false, af, false, bf, (short)0, acc3, false, false);
    }
    if (colB < NROWS) {
      #pragma unroll
      for (int dd = 0; dd < 4; ++dd) {
        int d = d0 + dd;
        v8f acc = (dd == 0) ? acc0 : (dd == 1) ? acc1 : (dd == 2) ? acc2 : acc3;
        #pragma unroll
        for (int r = 0; r < 8; ++r) {
          int rowA = rBase + r;
          if (rowA < NROWS) {
            float cr = acc[r];
            float l2 = sq[rowA * DF + d] + sq[colB * DF + d] - 2.f * cr;
            float kk = __expf(-l2 * inv0) + __expf(-l2 * inv1) + __expf(-l2 * inv2)
                     + __expf(-l2 * inv3) + __expf(-l2 * inv4);
            part += (((rowA < NSUB) == (colB < NSUB)) ? kk : -kk);
          }
        }
      }
    }
  }
  float tot = block_reduce(part, red);
  if (tid == 0) mmd_out[b] = tot / (float)(NSUB * NSUB * DF);
}

// ---------------- Kernel C: scalar triplet/margin/intra/inter assembly ----------------
__global__ void finalize_kernel(const int* __restrict__ lens,
                                const float* __restrict__ sd,
                                const float* __restrict__ mmd,
                                float* __restrict__ out) {
  if (threadIdx.x != 0) return;
  float accum = 0.f;
  for (int w = 0; w < NWK; ++w) {
    float dsv[15], dtv[15];
    for (int q = 0; q < 15; ++q) {
      int a = STEPL * w, o = STEPL * w + 1 + q;
      float denom = (float)(lens[a] + lens[o]);
      int p = w * 15 + q;
      dsv[q] = sd[p] / denom;        // student
      dtv[q] = sd[60 + p] / denom;   // teacher
    }
    // dist_g = dsv[0..4], dist_n = dsv[5..14]
    float mx = 0.f;
    float dtt[10][5];
    for (int n = 0; n < 10; ++n)
      for (int g = 0; g < 5; ++g) {
        float v = dtv[5 + n] - dtv[g];
        v = v > 0.f ? v : 0.f;
        dtt[n][g] = v;
        mx = v > mx ? v : mx;
      }
    float lsum = 0.f; int nz = 0;
    float fscale = 0.9f / mx;                      // (MARGIN_MAX-MARGIN_MIN)/mx
    for (int n = 0; n < 10; ++n)
      for (int g = 0; g < 5; ++g) {
        float fd = fscale * dtt[n][g] + 0.1f;      // + MARGIN_MIN
        float lk = dsv[g] - dsv[5 + n] + fd;
        lk = lk > 0.f ? lk : 0.f;
        if (lk != 0.f) nz++;
        lsum += lk;
      }
    float lv = lsum / ((float)nz + 1.f);
    float ca = 0.f, cb = 0.f, intra = 0.f;
    for (int g = 0; g < 5; ++g) ca += dsv[g];
    ca *= 0.2f;
    for (int n = 0; n < 5; ++n) cb += dsv[5 + n];
    cb *= 0.2f;
    for (int g = 0; g < 5; ++g) intra += dsv[g] - ca;
    float dd = fabsf(ca - cb);
    float inter = 1.0f - dd;                       // BETA - |ca-cb|
    if (inter < 0.f) inter = 0.f;
    accum += lv + intra * 0.1f + inter * 0.1f;     // P_W = R_W = 0.1
  }
  float total_loss = accum * 0.25f;                // mean over NW
  float mmax = 0.f;                                // mmds slot 9 is 0
  for (int k = 0; k < NPAIRS; ++k) mmax = mmd[k] > mmax ? mmd[k] : mmax;
  out[0] = total_loss + 0.5f * mmax;               // ALPHA = 0.5
}

// ---------------- launch ----------------
extern "C" void kernel_launch(void* const* d_in, const int* in_sizes, int n_in,
                              void* d_out, int out_size, void* d_ws, size_t ws_size,
                              hipStream_t stream) {
  (void)in_sizes; (void)n_in; (void)out_size; (void)ws_size;
  const float* data    = (const float*)d_in[0];
  const int*   lens    = (const int*)d_in[1];
  const float* teacher = (const float*)d_in[2];
  float* out = (float*)d_out;
  float* sd  = (float*)d_ws;        // 120 floats
  float* mmd = sd + 128;            // 9 floats

  diag_dtw_kernel<<<120, 256, 0, stream>>>(data, teacher, sd);
  mmd_kernel<<<NPAIRS, 128, 0, stream>>>(data, mmd);
  finalize_kernel<<<1, 64, 0, stream>>>(lens, sd, mmd, out);
}